// BipartiteMPNN_69243462746220
// MI455X (gfx1250) — compile-verified
//
#include <hip/hip_runtime.h>
#include <hip/hip_bf16.h>
#include <stdint.h>

typedef __attribute__((ext_vector_type(2))) float v2f;
typedef __attribute__((ext_vector_type(8))) float v8f;

#define HD 128          // feature dim H
#define K1 256          // 2*H
#define NB 16           // batch count B
#define ROWS 32         // rows per workgroup (2 M-tiles)
#define APAD 260        // LDS row stride in floats (260 mod 64 == 4 -> conflict-free)
#define INF_BITS 0x7F800000u

// ---------------------------------------------------------------------------
// Pack W[k][n] (row-major, K=256) into WMMA-fragment order:
//   P[((k>>2)*2 + ((k>>1)&1)) * N + n] is a v2f = {W[k&~1][n], W[k|1][n]}
// so each lane's B fragment is ONE global_load_b64, coalesced across lanes.
// ---------------------------------------------------------------------------
__global__ void pack_w_kernel(const float* __restrict__ W, float* __restrict__ P, int N) {
    int i = blockIdx.x * blockDim.x + threadIdx.x;
    if (i >= K1 * N) return;
    int k = i / N, n = i - k * N;
    int g2 = (k >> 2) * 2 + ((k >> 1) & 1);
    int lo = k & 1;
    P[((size_t)g2 * N + n) * 2 + lo] = W[i];
}

// ---------------------------------------------------------------------------
// Fused 2-layer MLP: out = relu(relu([A0|A1] @ W1 + b1) @ W2 + b2)
// W1p/W2p are fragment-packed (see above). 32 rows per WG, 128 threads = 4 waves.
// One 33KB LDS tile buffer, reused for input tile then hidden tile.
// A tile staged with GLOBAL_LOAD_ASYNC_TO_LDS_B128 (ASYNCcnt path).
// ---------------------------------------------------------------------------
__global__ __launch_bounds__(128) void mlp2_kernel(
    const float* __restrict__ A0, const float* __restrict__ A1,
    const float* __restrict__ W1p, const float* __restrict__ b1,
    const float* __restrict__ W2p, const float* __restrict__ b2,
    float* __restrict__ out, int nRows)
{
    __shared__ __align__(16) float As[ROWS][APAD];   // input tile, then hidden tile

    const int tid  = threadIdx.x;
    const int wv   = tid >> 5;
    const int lane = tid & 31;
    const int M    = lane & 15;     // frag row (A) / frag col N (B, C/D)
    const int kh   = lane >> 4;     // K-half selector per ISA 16x4 layouts
    const int r0   = blockIdx.x * ROWS;

    const unsigned ldsBase = (unsigned)(size_t)&As[0][0];   // low 32 bits = LDS offset

    // ---- stage [A0 | A1] tile via async B128 copies (uniform saddr per loop)
    for (int i = tid; i < ROWS * 32; i += 128) {            // 32 vec4 per 128-col half
        int row = i >> 5;
        int col = (i & 31) * 4;
        int gr = r0 + row; if (gr >= nRows) gr = nRows - 1;
        unsigned voff  = (unsigned)(((size_t)gr * HD + col) * 4);
        unsigned laddr = ldsBase + (unsigned)((row * APAD + col) * 4);
        asm volatile("global_load_async_to_lds_b128 %0, %1, %2"
                     :: "v"(laddr), "v"(voff), "s"(A0) : "memory");
    }
    for (int i = tid; i < ROWS * 32; i += 128) {
        int row = i >> 5;
        int col = (i & 31) * 4;
        int gr = r0 + row; if (gr >= nRows) gr = nRows - 1;
        unsigned voff  = (unsigned)(((size_t)gr * HD + col) * 4);
        unsigned laddr = ldsBase + (unsigned)((row * APAD + HD + col) * 4);
        asm volatile("global_load_async_to_lds_b128 %0, %1, %2"
                     :: "v"(laddr), "v"(voff), "s"(A1) : "memory");
    }
    asm volatile("s_wait_asynccnt 0x0" ::: "memory");
    __syncthreads();

    const v2f* W1v = (const v2f*)W1p;
    const v2f* W2v = (const v2f*)W2p;

    // ---- GEMM1: 32x256 tile; wave owns cols [wv*64, wv*64+64), 2 M-tiles
    v8f acc[2][4] = {};
    const int n0w = wv * 64;
    for (int kk = 0; kk < K1; kk += 4) {
        const int kb = kk + 2 * kh;                 // K = 2*kh + vgpr (ISA A/B layout)
        v2f a0 = *(const v2f*)&As[M][kb];           // ds_load_b64, conflict-free
        v2f a1 = *(const v2f*)&As[16 + M][kb];
        const int g2 = (kk >> 1) + kh;              // (kk>>2)*2 + kh
        #pragma unroll
        for (int t = 0; t < 4; ++t) {
            const int n = n0w + t * 16 + M;
            v2f b = W1v[(size_t)g2 * K1 + n];       // global_load_b64, coalesced
            acc[0][t] = __builtin_amdgcn_wmma_f32_16x16x4_f32(
                false, a0, false, b, (short)0, acc[0][t], false, false);
            acc[1][t] = __builtin_amdgcn_wmma_f32_16x16x4_f32(
                false, a1, false, b, (short)0, acc[1][t], false, false);
        }
    }
    __syncthreads();                                // everyone done reading As

    // bias + relu -> hidden tile (row-major [m][k'], overwrites As)
    #pragma unroll
    for (int t = 0; t < 4; ++t) {
        const int n = n0w + t * 16 + M;
        const float bb = b1[n];
        #pragma unroll
        for (int mt = 0; mt < 2; ++mt) {
            #pragma unroll
            for (int j = 0; j < 8; ++j) {
                const int m = mt * 16 + j + 8 * kh; // C/D layout: M = j + 8*(lane>=16)
                float v = acc[mt][t][j] + bb;
                As[m][n] = v > 0.f ? v : 0.f;
            }
        }
    }
    __syncthreads();

    // ---- GEMM2: 32x128 tile; wave owns cols [wv*32, wv*32+32), 2 M-tiles
    v8f acc2[2][2] = {};
    for (int kk = 0; kk < K1; kk += 4) {
        const int kb = kk + 2 * kh;
        v2f a0 = *(const v2f*)&As[M][kb];
        v2f a1 = *(const v2f*)&As[16 + M][kb];
        const int g2 = (kk >> 1) + kh;
        #pragma unroll
        for (int t = 0; t < 2; ++t) {
            const int n = wv * 32 + t * 16 + M;
            v2f b = W2v[(size_t)g2 * HD + n];
            acc2[0][t] = __builtin_amdgcn_wmma_f32_16x16x4_f32(
                false, a0, false, b, (short)0, acc2[0][t], false, false);
            acc2[1][t] = __builtin_amdgcn_wmma_f32_16x16x4_f32(
                false, a1, false, b, (short)0, acc2[1][t], false, false);
        }
    }
    #pragma unroll
    for (int t = 0; t < 2; ++t) {
        const int n = wv * 32 + t * 16 + M;
        const float bb = b2[n];
        #pragma unroll
        for (int mt = 0; mt < 2; ++mt) {
            #pragma unroll
            for (int j = 0; j < 8; ++j) {
                const int gr = r0 + mt * 16 + j + 8 * kh;
                if (gr < nRows) {
                    float v = acc2[mt][t][j] + bb;
                    out[(size_t)gr * HD + n] = v > 0.f ? v : 0.f;
                }
            }
        }
    }
}

// ---------------------------------------------------------------------------
__global__ void fill_u32_kernel(unsigned* __restrict__ p, unsigned v, int n) {
    int i = blockIdx.x * blockDim.x + threadIdx.x;
    if (i < n) p[i] = v;
}

// segment-min over edges: next_y[dst][h] = min(x_degree[src][h]).
// x_degree >= 0 (post-ReLU) so u32 atomicMin == float min on the bit pattern.
__global__ void scatter_min_kernel(
    const int* __restrict__ e_dst, const int* __restrict__ e_src,
    const unsigned char* __restrict__ e_mask,
    const float* __restrict__ xdeg, unsigned* __restrict__ y_u, int EH)
{
    int i = blockIdx.x * blockDim.x + threadIdx.x;
    if (i >= EH) return;
    int e = i >> 7;            // /128
    int h = i & (HD - 1);
    if (!e_mask[e]) return;
    unsigned v = __float_as_uint(xdeg[(size_t)e_src[e] * HD + h]);
    atomicMin(&y_u[(size_t)e_dst[e] * HD + h], v);
}

// replace +inf (empty segment) with 0.0, in place
__global__ void finalize_inf0_kernel(unsigned* __restrict__ p, int n) {
    int i = blockIdx.x * blockDim.x + threadIdx.x;
    if (i < n) { unsigned v = p[i]; p[i] = (v == INF_BITS) ? 0u : v; }
}

// per-batch min of next_y (masked). Block reduces 256 rows into LDS [16][128]
// (thread t owns feature column t exclusively -> race free), then 16 atomics/thread.
__global__ __launch_bounds__(128) void eps_min_kernel(
    const float* __restrict__ y, const unsigned char* __restrict__ y_mask,
    const int* __restrict__ bidx_y, unsigned* __restrict__ eps_u, int NY)
{
    __shared__ float lmin[NB * HD];
    const int h = threadIdx.x;
    #pragma unroll
    for (int b = 0; b < NB; ++b) lmin[b * HD + h] = __uint_as_float(INF_BITS);
    __syncthreads();
    const int r0 = blockIdx.x * 256;
    for (int r = 0; r < 256; ++r) {
        int row = r0 + r;
        if (row < NY && y_mask[row]) {
            int b = bidx_y[row] & (NB - 1);
            float v = y[(size_t)row * HD + h];
            float c = lmin[b * HD + h];
            lmin[b * HD + h] = v < c ? v : c;
        }
    }
    #pragma unroll
    for (int b = 0; b < NB; ++b)
        atomicMin(&eps_u[b * HD + h], __float_as_uint(lmin[b * HD + h]));
}

// one atomic per edge: counts[src][batch(dst)] += 1
__global__ void count_edges_kernel(
    const int* __restrict__ e_dst, const int* __restrict__ e_src,
    const unsigned char* __restrict__ e_mask, const int* __restrict__ bidx_y,
    float* __restrict__ counts, int E)
{
    int e = blockIdx.x * blockDim.x + threadIdx.x;
    if (e >= E || !e_mask[e]) return;
    int b = bidx_y[e_dst[e]] & (NB - 1);
    atomicAdd(&counts[(size_t)e_src[e] * NB + b], 1.0f);
}

// msg[x] = sum_b counts[x][b] * next_eps[b]   (segment_sum factorized)
__global__ __launch_bounds__(128) void expand_msg_kernel(
    const float* __restrict__ counts, const float* __restrict__ eps,
    float* __restrict__ msg, int NX)
{
    __shared__ float eps_s[NB * HD];
    __shared__ float cnt_s[32 * NB];
    const int h = threadIdx.x;
    const int x0 = blockIdx.x * 32;
    for (int i = h; i < NB * HD; i += 128) eps_s[i] = eps[i];
    for (int i = h; i < 32 * NB; i += 128) {
        int x = x0 + (i >> 4);
        cnt_s[i] = (x < NX) ? counts[(size_t)x * NB + (i & (NB - 1))] : 0.f;
    }
    __syncthreads();
    for (int xi = 0; xi < 32; ++xi) {
        int x = x0 + xi;
        if (x >= NX) break;
        float s = 0.f;
        #pragma unroll
        for (int b = 0; b < NB; ++b) s += cnt_s[xi * NB + b] * eps_s[b * HD + h];
        msg[(size_t)x * HD + h] = s;
    }
}

// ---------------------------------------------------------------------------
extern "C" void kernel_launch(void* const* d_in, const int* in_sizes, int n_in,
                              void* d_out, int out_size, void* d_ws, size_t ws_size,
                              hipStream_t stream)
{
    const float* h_x  = (const float*)d_in[0];
    const float* h_xd = (const float*)d_in[1];
    const float* W1m  = (const float*)d_in[2];
    const float* b1m  = (const float*)d_in[3];
    const float* W2m  = (const float*)d_in[4];
    const float* b2m  = (const float*)d_in[5];
    const float* W1u  = (const float*)d_in[6];
    const float* b1u  = (const float*)d_in[7];
    const float* W2u  = (const float*)d_in[8];
    const float* b2u  = (const float*)d_in[9];
    const int*   ei   = (const int*)d_in[10];
    const unsigned char* y_mask = (const unsigned char*)d_in[12];
    const unsigned char* e_mask = (const unsigned char*)d_in[13];
    const int*   bidx_y = (const int*)d_in[15];

    const int NX = in_sizes[0] / HD;
    const int NY = in_sizes[12];
    const int E  = in_sizes[13];
    const int B  = (out_size - NX * HD - NY * HD) / HD;   // == NB == 16

    const int* e_dst = ei;
    const int* e_src = ei + E;

    float* outX = (float*)d_out;
    float* outY = outX + (size_t)NX * HD;
    float* outE = outY + (size_t)NY * HD;

    float* xdeg = (float*)d_ws;                   // NX*HD floats; reused as msg later
    float* cnts = xdeg + (size_t)NX * HD;         // NX*NB floats
    float* pW1m = cnts + (size_t)NX * NB;         // packed weights
    float* pW2m = pW1m + (size_t)K1 * K1;
    float* pW1u = pW2m + (size_t)K1 * HD;
    float* pW2u = pW1u + (size_t)K1 * K1;

    const int TB = 256;

    // pack weights into WMMA-fragment order (b64 per B fragment)
    pack_w_kernel<<<(K1 * K1 + TB - 1) / TB, TB, 0, stream>>>(W1m, pW1m, K1);
    pack_w_kernel<<<(K1 * HD + TB - 1) / TB, TB, 0, stream>>>(W2m, pW2m, HD);
    pack_w_kernel<<<(K1 * K1 + TB - 1) / TB, TB, 0, stream>>>(W1u, pW1u, K1);
    pack_w_kernel<<<(K1 * HD + TB - 1) / TB, TB, 0, stream>>>(W2u, pW2u, HD);

    // init: next_y and next_eps regions to +inf bits, counts to zero
    fill_u32_kernel<<<(NY * HD + TB - 1) / TB, TB, 0, stream>>>((unsigned*)outY, INF_BITS, NY * HD);
    fill_u32_kernel<<<(B * HD + TB - 1) / TB, TB, 0, stream>>>((unsigned*)outE, INF_BITS, B * HD);
    fill_u32_kernel<<<(NX * NB + TB - 1) / TB, TB, 0, stream>>>((unsigned*)cnts, 0u, NX * NB);

    // x_degree = MLP_m([h_x | h_x_degree])
    mlp2_kernel<<<(NX + ROWS - 1) / ROWS, 128, 0, stream>>>(h_x, h_xd, pW1m, b1m, pW2m, b2m, xdeg, NX);

    // next_y = segment_min over edges; inf->0
    const int EH = E * HD;
    scatter_min_kernel<<<(EH + TB - 1) / TB, TB, 0, stream>>>(e_dst, e_src, e_mask, xdeg, (unsigned*)outY, EH);
    finalize_inf0_kernel<<<(NY * HD + TB - 1) / TB, TB, 0, stream>>>((unsigned*)outY, NY * HD);

    // next_eps = per-batch min of masked next_y; inf->0
    eps_min_kernel<<<(NY + 255) / 256, 128, 0, stream>>>(outY, y_mask, bidx_y, (unsigned*)outE, NY);
    finalize_inf0_kernel<<<(B * HD + TB - 1) / TB, TB, 0, stream>>>((unsigned*)outE, B * HD);

    // msg_y_to_x = segment_sum of next_eps[batch[dst]] over src  (count factorization)
    count_edges_kernel<<<(E + TB - 1) / TB, TB, 0, stream>>>(e_dst, e_src, e_mask, bidx_y, cnts, E);
    expand_msg_kernel<<<(NX + 31) / 32, 128, 0, stream>>>(cnts, outE, xdeg, NX);  // msg overwrites xdeg (dead)

    // next_x = MLP_u([h_x | msg])
    mlp2_kernel<<<(NX + ROWS - 1) / ROWS, 128, 0, stream>>>(h_x, xdeg, pW1u, b1u, pW2u, b2u, outX, NX);
}